// ConditionalOT_26036091748499
// MI455X (gfx1250) — compile-verified
//
#include <hip/hip_runtime.h>

typedef float v2f __attribute__((ext_vector_type(2)));
typedef float v8f __attribute__((ext_vector_type(8)));
typedef float f4  __attribute__((ext_vector_type(4)));
typedef unsigned int u32x4 __attribute__((ext_vector_type(4)));
typedef int i32x4 __attribute__((ext_vector_type(4)));
typedef int i32x8 __attribute__((ext_vector_type(8)));

#define N_ROWS  4096
#define D_LAT   512
#define D_CHEM  256
#define K_TOT   768
#define N_COLS  512
#define OT_EPS  0.1f
#define MAX_IT  50

// ---------------- Kernel A: fused GEMM + GELU + (h @ W2) -> c ----------------
#define BR       128   // rows per block (8 waves x 16 rows)
#define KC       32    // K panel staged in LDS
#define NC       128   // column chunk (8 WMMA tiles of 16)
#define APAD     2     // v2f padding for A panel
#define BSTRIDE  130   // 128 cols + 2 dwords TDM pad per row

__global__ __launch_bounds__(256) void k_gemm_gelu_cost(
    const float* __restrict__ zc, const float* __restrict__ pp,
    const float* __restrict__ W1, const float* __restrict__ b1,
    const float* __restrict__ W2, const float* __restrict__ b2,
    float* __restrict__ cvec)
{
    // A: K-pairs packed contiguously -> single ds_load_b64 per WMMA A operand
    __shared__ v2f   As2[KC / 2][BR + APAD];
    // B: filled by the Tensor Data Mover, row stride 130 dwords via TDM padding
    __shared__ float Bs[KC * BSTRIDE];

    const int tid   = threadIdx.x;
    const int wave  = tid >> 5;
    const int lane  = tid & 31;
    const int half  = lane >> 4;   // lane group: selects K pair / M+8
    const int l16   = lane & 15;
    const int rbase = blockIdx.x * BR;

    const unsigned lds_b = (unsigned)(uintptr_t)(void*)&Bs[0];

    float crow = 0.0f;   // per-row partial of gelu(h) . W2

    for (int nc = 0; nc < N_COLS / NC; ++nc) {
        v8f acc[8];
#pragma unroll
        for (int t = 0; t < 8; ++t) acc[t] = (v8f)0.0f;

        for (int kc = 0; kc < K_TOT / KC; ++kc) {
            __syncthreads();   // previous panel fully consumed

            // ---- issue TDM load of W1 panel: tile KC x NC of the 768x512 tensor
            if (wave == 0) {
                unsigned long long ga =
                    (unsigned long long)(uintptr_t)(&W1[(kc * KC) * N_COLS + nc * NC]);
                u32x4 g0;
                g0.x = 1u;                                   // count=1, no gather
                g0.y = lds_b;                                // LDS byte address
                g0.z = (unsigned)ga;                         // global addr lo
                g0.w = (unsigned)((ga >> 32) & 0x1FFFFFFull) // global addr hi (57b)
                       | 0x80000000u;                        // type=2 ("image")
                i32x8 g1;
                g1[0] = (2 << 16)        // data_size = 4 bytes
                      | (1 << 20)        // pad_enable
                      | (6 << 22)        // pad_interval: every 128 dwords
                      | (1 << 25);       // pad_amount: 2 dwords -> row stride 130
                g1[1] = (N_COLS & 0xFFFF) << 16;  // tensor_dim0[15:0]  (512)
                g1[2] = (K_TOT  & 0xFFFF) << 16;  // tensor_dim1[15:0]  (768)
                g1[3] = (NC << 16);               // tile_dim0 = 128
                g1[4] = KC;                       // tile_dim1 = 32
                g1[5] = N_COLS;                   // tensor_dim0_stride = 512
                g1[6] = 0;
                g1[7] = 0;
                i32x4 gz = (i32x4)0;
#if defined(__clang_major__) && __clang_major__ >= 23
                i32x8 g4 = (i32x8)0;
                __builtin_amdgcn_tensor_load_to_lds(g0, g1, gz, gz, g4, 0);
#else
                __builtin_amdgcn_tensor_load_to_lds(g0, g1, gz, gz, 0);
#endif
            }

            // ---- cooperative A staging: contiguous 8B global loads -> paired LDS
#pragma unroll
            for (int idx = tid; idx < (KC / 2) * BR; idx += 256) {
                int kp  = idx & 15;          // K pair index
                int row = idx >> 4;
                int gk  = kc * KC + 2 * kp;
                int gr  = rbase + row;
                v2f val = (gk < D_LAT)
                    ? *(const v2f*)(zc + (size_t)gr * D_LAT + gk)
                    : *(const v2f*)(pp + (size_t)gr * D_CHEM + (gk - D_LAT));
                As2[kp][row] = val;
            }

            if (kc + 1 < K_TOT / KC) {  // prefetch next A panel (64B/lane)
                int gk = (kc + 1) * KC;
                const float* na = (gk < D_LAT)
                    ? &zc[(size_t)(rbase + (tid >> 1)) * D_LAT + gk + (tid & 1) * 16]
                    : &pp[(size_t)(rbase + (tid >> 1)) * D_CHEM + (gk - D_LAT) + (tid & 1) * 16];
                __builtin_prefetch(na, 0, 1);
            }

            if (wave == 0) __builtin_amdgcn_s_wait_tensorcnt(0);
            __syncthreads();   // A (ds stores) + B (TDM) both visible

            // K = 2*(lane/16) + vgpr within each 4-wide chunk (ISA 16x4 layout)
#pragma unroll
            for (int kk = 0; kk < KC; kk += 4) {
                int ka = kk + 2 * half;
                v2f a = As2[ka >> 1][wave * 16 + l16];   // one ds_load_b64

                // B operands: one ds_load_2addr_b32 per tile loads the exact
                // (row ka, row ka+1) pair into an aligned VGPR pair -- no moves.
                // offset0 = t*16, offset1 = 130 + t*16 (8-bit element offsets).
                unsigned baddr = lds_b + (unsigned)((ka * BSTRIDE + l16) * 4);
                v2f b0, b1t, b2t, b3t, b4t, b5t, b6t, b7t;
                asm volatile(
                    "ds_load_2addr_b32 %0, %8 offset0:0   offset1:130\n\t"
                    "ds_load_2addr_b32 %1, %8 offset0:16  offset1:146\n\t"
                    "ds_load_2addr_b32 %2, %8 offset0:32  offset1:162\n\t"
                    "ds_load_2addr_b32 %3, %8 offset0:48  offset1:178\n\t"
                    "ds_load_2addr_b32 %4, %8 offset0:64  offset1:194\n\t"
                    "ds_load_2addr_b32 %5, %8 offset0:80  offset1:210\n\t"
                    "ds_load_2addr_b32 %6, %8 offset0:96  offset1:226\n\t"
                    "ds_load_2addr_b32 %7, %8 offset0:112 offset1:242\n\t"
                    "s_wait_dscnt 0x0"
                    : "=v"(b0), "=v"(b1t), "=v"(b2t), "=v"(b3t),
                      "=v"(b4t), "=v"(b5t), "=v"(b6t), "=v"(b7t)
                    : "v"(baddr));

                acc[0] = __builtin_amdgcn_wmma_f32_16x16x4_f32(
                    false, a, false, b0,  (short)0, acc[0], false, false);
                acc[1] = __builtin_amdgcn_wmma_f32_16x16x4_f32(
                    false, a, false, b1t, (short)0, acc[1], false, false);
                acc[2] = __builtin_amdgcn_wmma_f32_16x16x4_f32(
                    false, a, false, b2t, (short)0, acc[2], false, false);
                acc[3] = __builtin_amdgcn_wmma_f32_16x16x4_f32(
                    false, a, false, b3t, (short)0, acc[3], false, false);
                acc[4] = __builtin_amdgcn_wmma_f32_16x16x4_f32(
                    false, a, false, b4t, (short)0, acc[4], false, false);
                acc[5] = __builtin_amdgcn_wmma_f32_16x16x4_f32(
                    false, a, false, b5t, (short)0, acc[5], false, false);
                acc[6] = __builtin_amdgcn_wmma_f32_16x16x4_f32(
                    false, a, false, b6t, (short)0, acc[6], false, false);
                acc[7] = __builtin_amdgcn_wmma_f32_16x16x4_f32(
                    false, a, false, b7t, (short)0, acc[7], false, false);
            }
        }

        // Epilogue: bias + exact GELU + multiply by W2, reduce across columns.
        // C/D layout: vgpr r, lane half g, col l16 -> element (M=r+8g, N=t*16+l16)
#pragma unroll
        for (int t = 0; t < 8; ++t) {
            int col   = nc * NC + t * 16 + l16;
            float b1c = b1[col];
            float w2c = W2[col];
#pragma unroll
            for (int r = 0; r < 8; ++r) {
                float x = acc[t][r] + b1c;
                float g = 0.5f * x * (1.0f + erff(x * 0.70710678118654752f));
                float contrib = g * w2c;
#pragma unroll
                for (int m = 1; m < 16; m <<= 1)            // reduce within half
                    contrib += __shfl_xor(contrib, m, 32);
                if ((lane & 7) == r) crow += contrib;        // deterministic pickup
            }
        }
    }

    if ((lane & 15) < 8) {
        int grow = rbase + wave * 16 + half * 8 + (lane & 7);
        cvec[grow] = crow + b2[0];
    }
}

// ---------------- Kernel B: Sinkhorn (rank-1 cost -> O(N) per iteration) ----
__device__ __forceinline__ float blk_max(float x, float* red, int tid) {
    red[tid] = x; __syncthreads();
    for (int s = 512; s > 0; s >>= 1) {
        if (tid < s) red[tid] = fmaxf(red[tid], red[tid + s]);
        __syncthreads();
    }
    float r = red[0]; __syncthreads();
    return r;
}
__device__ __forceinline__ float blk_sum(float x, float* red, int tid) {
    red[tid] = x; __syncthreads();
    for (int s = 512; s > 0; s >>= 1) {
        if (tid < s) red[tid] += red[tid + s];
        __syncthreads();
    }
    float r = red[0]; __syncthreads();
    return r;
}

__global__ __launch_bounds__(1024) void k_sinkhorn(
    const float* __restrict__ cvec, float* __restrict__ avec, float* __restrict__ bvec)
{
    __shared__ float red[1024];
    const int tid = threadIdx.x;
    const float inv_eps  = 1.0f / OT_EPS;
    const float eps_logm = OT_EPS * logf(1.0f / (float)N_ROWS + 1e-8f);

    float u[4], v[4], cl[4];
#pragma unroll
    for (int k = 0; k < 4; ++k) {
        u[k] = 0.0f; v[k] = 0.0f;
        cl[k] = cvec[tid + k * 1024];
    }

    for (int it = 0; it < MAX_IT; ++it) {
        // lse_v = logsumexp(v/eps); u_i = eps*logm - (u_i - c_i) - eps*lse_v
        float lm = -3.0e38f;
#pragma unroll
        for (int k = 0; k < 4; ++k) lm = fmaxf(lm, v[k] * inv_eps);
        float m  = blk_max(lm, red, tid);
        float ls = 0.0f;
#pragma unroll
        for (int k = 0; k < 4; ++k) ls += expf(v[k] * inv_eps - m);
        float lse_v = m + logf(blk_sum(ls, red, tid));
#pragma unroll
        for (int k = 0; k < 4; ++k)
            u[k] = eps_logm - (u[k] - cl[k]) - OT_EPS * lse_v;

        // lse_u = logsumexp((u-c)/eps); v_j = eps*logm - v_j - eps*lse_u
        lm = -3.0e38f;
#pragma unroll
        for (int k = 0; k < 4; ++k) lm = fmaxf(lm, (u[k] - cl[k]) * inv_eps);
        m  = blk_max(lm, red, tid);
        ls = 0.0f;
#pragma unroll
        for (int k = 0; k < 4; ++k) ls += expf((u[k] - cl[k]) * inv_eps - m);
        float lse_u = m + logf(blk_sum(ls, red, tid));
#pragma unroll
        for (int k = 0; k < 4; ++k)
            v[k] = eps_logm - v[k] - OT_EPS * lse_u;
    }

#pragma unroll
    for (int k = 0; k < 4; ++k) {
        avec[tid + k * 1024] = expf((u[k] - cl[k]) * inv_eps);  // P = a_i * b_j
        bvec[tid + k * 1024] = expf(v[k] * inv_eps);
    }
}

// ---------------- Kernel C1: s = b^T @ z_perturbed (deterministic order) ----
__global__ __launch_bounds__(256) void k_colsum(
    const float* __restrict__ zp, const float* __restrict__ bvec,
    float* __restrict__ svec)
{
    int col = blockIdx.x * 256 + threadIdx.x;
    float acc = 0.0f;
    for (int j = 0; j < N_ROWS; j += 4) {
#pragma unroll
        for (int q = 0; q < 4; ++q)
            acc = fmaf(bvec[j + q], zp[(size_t)(j + q) * N_COLS + col], acc);
    }
    svec[col] = acc;
}

// ---------------- Kernel C2: z_ot = a_i * s ; P = a_i * b_j ----------------
__global__ __launch_bounds__(256) void k_outputs(
    const float* __restrict__ avec, const float* __restrict__ bvec,
    const float* __restrict__ svec, float* __restrict__ outZ,
    float* __restrict__ outP)
{
    const int i   = blockIdx.x;
    const int tid = threadIdx.x;
    const float ai = avec[i];

    const f4* s4 = (const f4*)svec;
    const f4* b4 = (const f4*)bvec;
    f4* z4 = (f4*)(outZ + (size_t)i * N_COLS);
    f4* p4 = (f4*)(outP + (size_t)i * N_ROWS);

    if (tid < N_COLS / 4) z4[tid] = ai * s4[tid];
#pragma unroll
    for (int k = 0; k < 4; ++k) {
        int idx = tid + k * 256;                 // 1024 f4 per P row
        __builtin_nontemporal_store(ai * b4[idx], &p4[idx]);
    }
}

// ---------------- host ----------------
extern "C" void kernel_launch(void* const* d_in, const int* in_sizes, int n_in,
                              void* d_out, int out_size, void* d_ws, size_t ws_size,
                              hipStream_t stream) {
    const float* zc = (const float*)d_in[0];   // z_control  (4096,512)
    const float* zp = (const float*)d_in[1];   // z_perturbed(4096,512)
    const float* pp = (const float*)d_in[2];   // p          (4096,256)
    const float* W1 = (const float*)d_in[3];   // (768,512)
    const float* b1 = (const float*)d_in[4];   // (512,)
    const float* W2 = (const float*)d_in[5];   // (512,1)
    const float* b2 = (const float*)d_in[6];   // (1,)

    float* ws   = (float*)d_ws;
    float* cvec = ws;            // 4096
    float* avec = ws + 4096;     // 4096
    float* bvec = ws + 8192;     // 4096
    float* svec = ws + 12288;    // 512

    float* outZ = (float*)d_out;                           // 4096*512
    float* outP = (float*)d_out + (size_t)N_ROWS * N_COLS; // 4096*4096

    k_gemm_gelu_cost<<<N_ROWS / BR, 256, 0, stream>>>(zc, pp, W1, b1, W2, b2, cvec);
    k_sinkhorn<<<1, 1024, 0, stream>>>(cvec, avec, bvec);
    k_colsum<<<N_COLS / 256, 256, 0, stream>>>(zp, bvec, svec);
    k_outputs<<<N_ROWS, 256, 0, stream>>>(avec, bvec, svec, outZ, outP);
}